// GATLayer_3418793968188
// MI455X (gfx1250) — compile-verified
//
#include <hip/hip_runtime.h>
#include <math.h>
#include <stdint.h>

// MI455X GAT layer: H = X@W ; e = s[i]+t[j] ; A = softmax(lrelu(e)) ; out = A@H
// Output = (out [8192x256], A [8192x8192]) concatenated in d_out.
// Bandwidth-bound on writing A (268 MB) -> fuse A production with A@H WMMA.
// Full-precision path: V_WMMA_F32_16X16X4_F32 for both GEMMs.
// H chunks stream L2 -> LDS via GLOBAL_LOAD_ASYNC_TO_LDS_B128, double-buffered
// so chunk c+1's DMA overlaps chunk c's exp + WMMA work.

#define NROWS   8192
#define IN_DIM  512
#define ODIM    256

typedef float v2f __attribute__((ext_vector_type(2)));
typedef float v8f __attribute__((ext_vector_type(8)));

__device__ __forceinline__ float lrelu(float x) { return fmaxf(x, 0.2f * x); }

// ---------------------------------------------------------------------------
// Kernel 1: H = X @ W   (M=8192, K=512, N=256), one 16x16 tile per wave.
// ---------------------------------------------------------------------------
__global__ __launch_bounds__(256) void gemm_h_kernel(const float* __restrict__ X,
                                                     const float* __restrict__ W,
                                                     float* __restrict__ H) {
    const int lane  = threadIdx.x & 31;
    const int wave  = threadIdx.x >> 5;
    const int gwave = blockIdx.x * 8 + wave;       // 8192 waves total
    const int m0 = (gwave >> 4) * 16;              // 512 row tiles
    const int n0 = (gwave & 15) * 16;              // 16  col tiles
    const int hi = lane >> 4;
    const int r  = lane & 15;

    const float* xrow = X + (m0 + r) * IN_DIM;     // A row for this lane
    const float* wcol = W + n0 + r;                // B column for this lane

    v8f acc = {};
    for (int k = 0; k < IN_DIM; k += 4) {
        const int kb = k + 2 * hi;
        v2f a = *(const v2f*)(xrow + kb);          // X[m][kb], X[m][kb+1]
        v2f b;
        b.x = wcol[kb * ODIM];                     // W[kb][n]
        b.y = wcol[(kb + 1) * ODIM];               // W[kb+1][n]
        acc = __builtin_amdgcn_wmma_f32_16x16x4_f32(false, a, false, b,
                                                    (short)0, acc, false, false);
    }
#pragma unroll
    for (int v = 0; v < 8; ++v)                    // C/D: VGPR v -> row v / v+8
        H[(m0 + v + 8 * hi) * ODIM + n0 + r] = acc[v];
}

// ---------------------------------------------------------------------------
// Kernel 2: s[i] = H[i,:] . a[0:256],  t[i] = H[i,:] . a[256:512]
// ---------------------------------------------------------------------------
__global__ __launch_bounds__(256) void st_kernel(const float* __restrict__ H,
                                                 const float* __restrict__ av,
                                                 float* __restrict__ s,
                                                 float* __restrict__ t) {
    const int lane = threadIdx.x & 31;
    const int row  = blockIdx.x * 8 + (threadIdx.x >> 5);
    const float* h = H + row * ODIM;
    float ss = 0.f, tt = 0.f;
    for (int k = lane; k < ODIM; k += 32) {
        float hv = h[k];
        ss += hv * av[k];
        tt += hv * av[ODIM + k];
    }
    for (int o = 16; o > 0; o >>= 1) {
        ss += __shfl_xor(ss, o, 32);
        tt += __shfl_xor(tt, o, 32);
    }
    if (lane == 0) { s[row] = ss; t[row] = tt; }
}

// ---------------------------------------------------------------------------
// Kernel 3: maxT = max_j t[j]   (single workgroup reduction)
// ---------------------------------------------------------------------------
__global__ __launch_bounds__(1024) void maxt_kernel(const float* __restrict__ t,
                                                    float* __restrict__ mx) {
    __shared__ float red[1024];
    const int tid = threadIdx.x;
    float m = -INFINITY;
    for (int i = tid; i < NROWS; i += 1024) m = fmaxf(m, t[i]);
    red[tid] = m;
    __syncthreads();
    for (int o = 512; o > 0; o >>= 1) {
        if (tid < o) red[tid] = fmaxf(red[tid], red[tid + o]);
        __syncthreads();
    }
    if (tid == 0) mx[0] = red[0];
}

// ---------------------------------------------------------------------------
// Kernel 4: m_i = lrelu(s_i + maxT);  rinv[i] = 1 / sum_j exp(lrelu(s_i+t_j)-m_i)
// (lrelu is monotone, so the analytic row max is exact.)
// ---------------------------------------------------------------------------
__global__ __launch_bounds__(256) void rowsum_kernel(const float* __restrict__ s,
                                                     const float* __restrict__ t,
                                                     const float* __restrict__ mx,
                                                     float* __restrict__ rinv,
                                                     float* __restrict__ mArr) {
    const int lane = threadIdx.x & 31;
    const int row  = blockIdx.x * 8 + (threadIdx.x >> 5);
    const float si = s[row];
    const float mi = lrelu(si + mx[0]);
    float sum = 0.f;
    for (int j = lane; j < NROWS; j += 32)
        sum += __expf(lrelu(si + t[j]) - mi);
    for (int o = 16; o > 0; o >>= 1) sum += __shfl_xor(sum, o, 32);
    if (lane == 0) { rinv[row] = 1.0f / sum; mArr[row] = mi; }
}

// ---------------------------------------------------------------------------
// Kernel 5 (fused): per workgroup: 16 rows x full 256 out-dims.
// j loops in chunks of 32. H[j:j+32,:] is DMA'd into LDS with
// global_load_async_to_lds_b128 (ASYNCcnt), double-buffered: chunk c+1 is in
// flight while chunk c is consumed. Per chunk: compute normalized attention
// tile (write straight to A output + LDS), then WMMA-accumulate
// out += A_tile @ H_tile with v_wmma_f32_16x16x4_f32 (8 waves x 2 N-tiles).
// ---------------------------------------------------------------------------
#define HSTR 260   // 256 + pad (floats)
#define ASTR 36    // 32 + pad  (floats)
#define CHUNK 32
#define NCH  (NROWS / CHUNK)
#define HBUF_FLTS (CHUNK * HSTR)

// Issue the 8 async b128 loads (32 floats) this thread owns for one H chunk.
__device__ __forceinline__ void async_load_chunk(const float* __restrict__ H,
                                                 int j0, uint32_t ldsBase, int tid) {
#pragma unroll
    for (int it = 0; it < 8; ++it) {
        const int fl = it * 1024 + tid * 4;        // 8192 floats per chunk
        const int j = fl >> 8, n = fl & 255;
        const uint32_t lds_off = ldsBase + (uint32_t)((j * HSTR + n) * 4);
        const uint64_t gaddr =
            (uint64_t)(uintptr_t)(H + (size_t)(j0 + j) * ODIM + n);
        asm volatile("global_load_async_to_lds_b128 %0, %1, off"
                     :: "v"(lds_off), "v"(gaddr) : "memory");
    }
}

__global__ __launch_bounds__(256) void attn_kernel(const float* __restrict__ H,
                                                   const float* __restrict__ s,
                                                   const float* __restrict__ tg,
                                                   const float* __restrict__ mArr,
                                                   const float* __restrict__ rinv,
                                                   float* __restrict__ Aout,
                                                   float* __restrict__ out) {
    __shared__ __align__(16) float Hs[2][HBUF_FLTS];  // ~65 KB double buffer
    __shared__ float As[16 * ASTR];                   // attention tile [16 x 32]
    __shared__ float sS[16], mS[16], rS[16];

    const int tid  = threadIdx.x;
    const int lane = tid & 31;
    const int wave = tid >> 5;
    const int hi   = lane >> 4;
    const int r    = lane & 15;
    const int m0   = blockIdx.x * 16;              // 512 workgroups

    if (tid < 16) {
        sS[tid] = s[m0 + tid];
        mS[tid] = mArr[m0 + tid];
        rS[tid] = rinv[m0 + tid];
    }

    const int nc0 = wave * 32;                     // this wave's 2 N-tiles
    const int nc1 = nc0 + 16;
    const int row2 = (tid * 2) >> 5;               // exp-tile assignment
    const int jj2  = (tid * 2) & 31;

    // LDS byte offsets of the two H buffers (low 32 bits of generic pointer).
    const uint32_t hsBase0 = (uint32_t)(uintptr_t)&Hs[0][0];
    const uint32_t hsBase1 = (uint32_t)(uintptr_t)&Hs[1][0];

    v8f acc0 = {}, acc1 = {};

    // Prologue: start DMA of chunk 0 into buffer 0.
    async_load_chunk(H, 0, hsBase0, tid);

    for (int c = 0; c < NCH; ++c) {
        const int j0 = c * CHUNK;
        const float* Hbuf = (c & 1) ? &Hs[1][0] : &Hs[0][0];

        __syncthreads();   // everyone done reading buf^1 (chunk c-1's WMMA)

        // Kick off DMA of chunk c+1 into the other buffer (overlaps below).
        if (c + 1 < NCH)
            async_load_chunk(H, j0 + CHUNK, (c & 1) ? hsBase0 : hsBase1, tid);

        // Normalized attention weights for this 16x32 tile (2 per thread);
        // runs while both DMAs are in flight.
        {
            const float si = sS[row2];
            const float mi = mS[row2];
            const float ri = rS[row2];
            const float p0 = __expf(lrelu(si + tg[j0 + jj2])     - mi) * ri;
            const float p1 = __expf(lrelu(si + tg[j0 + jj2 + 1]) - mi) * ri;
            As[row2 * ASTR + jj2]     = p0;
            As[row2 * ASTR + jj2 + 1] = p1;
            Aout[(size_t)(m0 + row2) * NROWS + j0 + jj2]     = p0;
            Aout[(size_t)(m0 + row2) * NROWS + j0 + jj2 + 1] = p1;
        }

        // Async loads complete in order per wave: waiting until <=8 remain
        // guarantees chunk c's 8 loads landed while chunk c+1's stay in flight.
        if (c + 1 < NCH) asm volatile("s_wait_asynccnt 0x8" ::: "memory");
        else             asm volatile("s_wait_asynccnt 0x0" ::: "memory");

        __syncthreads();   // all waves' chunk-c data + As tile visible

        // out[16 x 256] += A_tile[16 x 32] @ H_tile[32 x 256]
#pragma unroll
        for (int kk = 0; kk < 8; ++kk) {
            const int k = kk * 4 + 2 * hi;
            v2f a;
            a.x = As[r * ASTR + k];
            a.y = As[r * ASTR + k + 1];
            v2f b0, b1;
            b0.x = Hbuf[k * HSTR + nc0 + r];
            b0.y = Hbuf[(k + 1) * HSTR + nc0 + r];
            b1.x = Hbuf[k * HSTR + nc1 + r];
            b1.y = Hbuf[(k + 1) * HSTR + nc1 + r];
            acc0 = __builtin_amdgcn_wmma_f32_16x16x4_f32(false, a, false, b0,
                                                         (short)0, acc0, false, false);
            acc1 = __builtin_amdgcn_wmma_f32_16x16x4_f32(false, a, false, b1,
                                                         (short)0, acc1, false, false);
        }
    }

#pragma unroll
    for (int v = 0; v < 8; ++v) {
        out[(size_t)(m0 + v + 8 * hi) * ODIM + nc0 + r] = acc0[v];
        out[(size_t)(m0 + v + 8 * hi) * ODIM + nc1 + r] = acc1[v];
    }
}

// ---------------------------------------------------------------------------
extern "C" void kernel_launch(void* const* d_in, const int* in_sizes, int n_in,
                              void* d_out, int out_size, void* d_ws, size_t ws_size,
                              hipStream_t stream) {
    const float* X  = (const float*)d_in[0];   // [8192, 512]
    const float* W  = (const float*)d_in[1];   // [512, 256]
    const float* av = (const float*)d_in[2];   // [512]

    float* out = (float*)d_out;                // [8192, 256]
    float* A   = out + (size_t)NROWS * ODIM;   // [8192, 8192]

    // workspace carve-out (~8.2 MB)
    float* H    = (float*)d_ws;                // [8192, 256]
    float* s    = H + (size_t)NROWS * ODIM;
    float* t    = s + NROWS;
    float* rinv = t + NROWS;
    float* mArr = rinv + NROWS;
    float* mx   = mArr + NROWS;

    gemm_h_kernel <<<1024, 256, 0, stream>>>(X, W, H);
    st_kernel     <<<1024, 256, 0, stream>>>(H, av, s, t);
    maxt_kernel   <<<1,   1024, 0, stream>>>(t, mx);
    rowsum_kernel <<<1024, 256, 0, stream>>>(s, t, mx, rinv, mArr);
    attn_kernel   <<<512,  256, 0, stream>>>(H, s, t, mArr, rinv, A, out);
}